// poe_criterion_38044820308281
// MI455X (gfx1250) — compile-verified
//
#include <hip/hip_runtime.h>
#include <hip/hip_bf16.h>
#include <math.h>

#define NN 8192
#define MM 4
#define LL 256
#define NH 5           // hazard columns actually used: joint + 4 modality

typedef __attribute__((ext_vector_type(2))) float v2f;
typedef __attribute__((ext_vector_type(8))) float v8f;

// ---------------------------------------------------------------------------
// Kernel 1: Et[c*N + j] = exp(hazard_c[j]),  c=0 joint, c=1..4 modality rows
// ---------------------------------------------------------------------------
__global__ void __launch_bounds__(256)
k_prep_exp(const float* __restrict__ jh, const float* __restrict__ mh,
           float* __restrict__ Et) {
  int idx = blockIdx.x * blockDim.x + threadIdx.x;
  if (idx >= NH * NN) return;
  int c = idx / NN;
  int j = idx - c * NN;
  float h = (c == 0) ? jh[j] : mh[(c - 1) * NN + j];
  Et[idx] = expf(h);
}

// ---------------------------------------------------------------------------
// Kernel 2: risk[c][i] = sum_j (t[j] >= t[i]) * Et[c][j]  via f32 WMMA
// One wave per 16-row tile of i; K marched 4 at a time.
// ---------------------------------------------------------------------------
__global__ void __launch_bounds__(256)
k_risk_wmma(const float* __restrict__ target,   // (N,2) time,event interleaved
            const float* __restrict__ Et,       // (NH x N)
            float* __restrict__ risk) {         // (NH x N)
  __shared__ float s_time[NN];                  // 32 KB of the 320 KB WGP LDS
  for (int i = threadIdx.x; i < NN; i += 256)
    s_time[i] = target[2 * i];
  __syncthreads();

  const int wave   = threadIdx.x >> 5;          // 8 waves / block
  const int lane   = threadIdx.x & 31;
  const int lmod   = lane & 15;
  const int lhalf  = lane >> 4;                 // 0: K=0,1  1: K=2,3
  const int klo    = lhalf * 2;
  const int i_base = (blockIdx.x * 8 + wave) * 16;

  const float t_i = s_time[i_base + lmod];
  const int   c   = (lmod < NH) ? lmod : (NH - 1);   // clamp unused columns
  const float* __restrict__ Erow = Et + c * NN;

  v8f acc = {};
  #pragma unroll 4
  for (int j = 0; j < NN; j += 4) {
    // A tile (mask): lane holds K = klo, klo+1 for row M = lmod
    v2f a;
    a.x = (s_time[j + klo]     >= t_i) ? 1.0f : 0.0f;
    a.y = (s_time[j + klo + 1] >= t_i) ? 1.0f : 0.0f;
    // B tile: lane holds E[K=klo..klo+1][N=lmod]  (contiguous pair, 8B aligned)
    v2f b = *(const v2f*)(Erow + j + klo);
    acc = __builtin_amdgcn_wmma_f32_16x16x4_f32(
        /*neg_a=*/false, a, /*neg_b=*/false, b,
        /*c_mod=*/(short)0, acc, /*reuse_a=*/false, /*reuse_b=*/false);
  }

  // D layout: VGPR v = row (v + 8*lhalf), column = lmod
  if (lmod < NH) {
    const int row0 = i_base + lhalf * 8;
    #pragma unroll
    for (int v = 0; v < 8; ++v)
      risk[lmod * NN + row0 + v] = acc[v];
  }
}

// ---------------------------------------------------------------------------
// Kernel 3: cox numerators S_c = sum_i ev[i]*(h_c[i]-log(risk_c[i])), n_events
// Single block, fixed-order tree reduction -> deterministic.
// ---------------------------------------------------------------------------
__global__ void __launch_bounds__(256)
k_cox(const float* __restrict__ target, const float* __restrict__ jh,
      const float* __restrict__ mh, const float* __restrict__ risk,
      float* __restrict__ out6) {
  __shared__ float red[256];
  float s[NH + 1] = {0.f, 0.f, 0.f, 0.f, 0.f, 0.f};
  for (int i = threadIdx.x; i < NN; i += 256) {
    float ev = target[2 * i + 1];
    s[NH] += ev;
    if (ev != 0.f) {
      s[0] += ev * (jh[i] - logf(risk[i]));
      #pragma unroll
      for (int m = 0; m < MM; ++m)
        s[1 + m] += ev * (mh[m * NN + i] - logf(risk[(1 + m) * NN + i]));
    }
  }
  for (int q = 0; q < NH + 1; ++q) {
    red[threadIdx.x] = s[q];
    __syncthreads();
    for (int off = 128; off > 0; off >>= 1) {
      if (threadIdx.x < off) red[threadIdx.x] += red[threadIdx.x + off];
      __syncthreads();
    }
    if (threadIdx.x == 0) out6[q] = red[0];
    __syncthreads();
  }
}

// ---------------------------------------------------------------------------
// Kernel 4: KL partial sums (bandwidth-bound: ~80MB -> ~3.4us at 23.3 TB/s)
// float4 loads, grid-strided, per-block partials.
// ---------------------------------------------------------------------------
__device__ __forceinline__ float kl4(float4 l, float4 sc) {
  float r = 0.f;
  r += -logf(sc.x) + 0.5f * (sc.x * sc.x + l.x * l.x - 1.0f);
  r += -logf(sc.y) + 0.5f * (sc.y * sc.y + l.y * l.y - 1.0f);
  r += -logf(sc.z) + 0.5f * (sc.z * sc.z + l.z * l.z - 1.0f);
  r += -logf(sc.w) + 0.5f * (sc.w * sc.w + l.w * l.w - 1.0f);
  return r;
}

__global__ void __launch_bounds__(256)
k_kl(const float* __restrict__ jloc, const float* __restrict__ jscale,
     const float* __restrict__ mloc, const float* __restrict__ mscale,
     float* __restrict__ pj, float* __restrict__ pm) {
  const int JT4 = (NN * LL) / 4;        // 524288 float4
  const int MT4 = (MM * NN * LL) / 4;   // 2097152 float4
  const int nth = gridDim.x * blockDim.x;
  const int tid = blockIdx.x * blockDim.x + threadIdx.x;

  const float4* jl = (const float4*)jloc;
  const float4* js = (const float4*)jscale;
  const float4* ml = (const float4*)mloc;
  const float4* ms = (const float4*)mscale;

  float sj = 0.f, sm = 0.f;
  for (int i = tid; i < JT4; i += nth) sj += kl4(jl[i], js[i]);
  for (int i = tid; i < MT4; i += nth) sm += kl4(ml[i], ms[i]);

  __shared__ float red[256];
  red[threadIdx.x] = sj;
  __syncthreads();
  for (int off = 128; off > 0; off >>= 1) {
    if (threadIdx.x < off) red[threadIdx.x] += red[threadIdx.x + off];
    __syncthreads();
  }
  if (threadIdx.x == 0) pj[blockIdx.x] = red[0];
  __syncthreads();
  red[threadIdx.x] = sm;
  __syncthreads();
  for (int off = 128; off > 0; off >>= 1) {
    if (threadIdx.x < off) red[threadIdx.x] += red[threadIdx.x + off];
    __syncthreads();
  }
  if (threadIdx.x == 0) pm[blockIdx.x] = red[0];
}

// ---------------------------------------------------------------------------
// Kernel 5: finalize scalar
// ---------------------------------------------------------------------------
__global__ void __launch_bounds__(256)
k_final(const float* __restrict__ pj, const float* __restrict__ pm, int nparts,
        const float* __restrict__ out6,
        const float* __restrict__ alpha_p, const float* __restrict__ beta_p,
        float* __restrict__ out) {
  __shared__ float red[256];
  float sj = 0.f, sm = 0.f;
  for (int i = threadIdx.x; i < nparts; i += 256) { sj += pj[i]; sm += pm[i]; }
  red[threadIdx.x] = sj;
  __syncthreads();
  for (int off = 128; off > 0; off >>= 1) {
    if (threadIdx.x < off) red[threadIdx.x] += red[threadIdx.x + off];
    __syncthreads();
  }
  sj = red[0];
  __syncthreads();
  red[threadIdx.x] = sm;
  __syncthreads();
  for (int off = 128; off > 0; off >>= 1) {
    if (threadIdx.x < off) red[threadIdx.x] += red[threadIdx.x + off];
    __syncthreads();
  }
  sm = red[0];
  if (threadIdx.x == 0) {
    const float alpha = alpha_p[0], beta = beta_p[0];
    const float ne = out6[NH];
    const float cox_joint   = -out6[0] / ne;
    const float cox_mod_sum = -(out6[1] + out6[2] + out6[3] + out6[4]) / ne;
    const float joint_kl    = sj / (float)NN;
    const float mod_kl      = sm / (float)NN;
    out[0] = cox_joint + beta * joint_kl + alpha * (cox_mod_sum + beta * mod_kl);
  }
}

// ---------------------------------------------------------------------------
extern "C" void kernel_launch(void* const* d_in, const int* in_sizes, int n_in,
                              void* d_out, int out_size, void* d_ws, size_t ws_size,
                              hipStream_t stream) {
  const float* jh     = (const float*)d_in[0];   // (N)
  const float* mh     = (const float*)d_in[1];   // (M,N)
  const float* jloc   = (const float*)d_in[2];   // (N,L)
  const float* jscale = (const float*)d_in[3];   // (N,L)
  const float* mloc   = (const float*)d_in[4];   // (M,N,L)
  const float* mscale = (const float*)d_in[5];   // (M,N,L)
  const float* target = (const float*)d_in[6];   // (N,2)
  const float* alpha  = (const float*)d_in[7];
  const float* beta   = (const float*)d_in[8];
  float* out = (float*)d_out;

  float* ws   = (float*)d_ws;
  float* Et   = ws;                        // NH*N
  float* risk = ws + (size_t)NH * NN;      // NH*N
  float* out6 = risk + (size_t)NH * NN;    // 16
  float* pj   = out6 + 16;                 // 1024
  float* pm   = pj + 1024;                 // 1024

  const int KLBLOCKS = 1024;

  k_prep_exp<<<(NH * NN + 255) / 256, 256, 0, stream>>>(jh, mh, Et);
  k_risk_wmma<<<NN / 16 / 8, 256, 0, stream>>>(target, Et, risk);
  k_cox<<<1, 256, 0, stream>>>(target, jh, mh, risk, out6);
  k_kl<<<KLBLOCKS, 256, 0, stream>>>(jloc, jscale, mloc, mscale, pj, pm);
  k_final<<<1, 256, 0, stream>>>(pj, pm, KLBLOCKS, out6, alpha, beta, out);
}